// DetectionLoss_76656576299401
// MI455X (gfx1250) — compile-verified
//
#include <hip/hip_runtime.h>
#include <hip/hip_bf16.h>
#include <math.h>

typedef float v2f __attribute__((ext_vector_type(2)));
typedef float v8f __attribute__((ext_vector_type(8)));

#define NTHR   256
#define LAMBD  0.5f

// Per-item: cross-entropy contribution + confusion-matrix one-hot counts.
// ce_i = logsumexp(x0,x1) - x[label] = max(d,0) + log1p(exp(-|d|)),  d = x_other - x_label
__device__ __forceinline__ void acc_item(float x0, float x1, int lb,
                                         float& ce, float& c00, float& c01,
                                         float& c10, float& c11) {
    int pred = (x1 > x0) ? 1 : 0;              // argmax, ties -> 0 (matches jnp.argmax)
    float xl = lb ? x1 : x0;
    float xo = lb ? x0 : x1;
    float d  = xo - xl;
    ce += fmaxf(d, 0.f) + log1pf(expf(-fabsf(d)));
    int c = (lb << 1) | pred;                  // label*2 + pred
    c00 += (float)(c == 0);
    c01 += (float)(c == 1);
    c10 += (float)(c == 2);
    c11 += (float)(c == 3);
}

// Main pass: stream outputs/labels, accumulate 5 channels per thread,
// block-reduce via chained V_WMMA_F32_16X16X4_F32 (A=ones -> column sums),
// write 8 per-block partials (deterministic, no atomics).
__global__ __launch_bounds__(NTHR)
void det_loss_main(const float4* __restrict__ out4, const int2* __restrict__ lab2,
                   float* __restrict__ partials, long long pairs) {
    const int tid = threadIdx.x;
    const long long gid    = (long long)blockIdx.x * NTHR + tid;
    const long long stride = (long long)gridDim.x * NTHR;

    float ce = 0.f, c00 = 0.f, c01 = 0.f, c10 = 0.f, c11 = 0.f;

    for (long long i = gid; i < pairs; i += stride) {
        if (i + stride < pairs) {
            __builtin_prefetch((const void*)(out4 + i + stride), 0, 1);
            __builtin_prefetch((const void*)(lab2 + i + stride), 0, 1);
        }
        float4 o = out4[i];      // two items, 16B coalesced
        int2   l = lab2[i];      // two labels,  8B coalesced
        acc_item(o.x, o.y, l.x, ce, c00, c01, c10, c11);
        acc_item(o.z, o.w, l.y, ce, c00, c01, c10, c11);
    }

    // Stage partials channel-major in LDS: sm[ch*256 + tid]
    __shared__ float sm[8 * NTHR];
    sm[0 * NTHR + tid] = ce;
    sm[1 * NTHR + tid] = c00;
    sm[2 * NTHR + tid] = c01;
    sm[3 * NTHR + tid] = c10;
    sm[4 * NTHR + tid] = c11;
    sm[5 * NTHR + tid] = 0.f;
    sm[6 * NTHR + tid] = 0.f;
    sm[7 * NTHR + tid] = 0.f;
    __syncthreads();

    // Wave 0 (full EXEC) reduces all 8 channels x 256 values with 32 chained
    // f32 WMMAs. A = ones(16x4) => D[m][n] accumulates the column sums of B.
    // Column n: channel n&7, half (n>>3) of the 256 thread values. Column n's
    // 4 B-elements per step live in lanes {n, n+16} x {VGPR0, VGPR1}; the sum
    // is invariant to the hardware's K-ordering within the column.
    if (tid < 32) {
        const int n    = tid & 15;
        const int ch   = n & 7;
        const int half = n >> 3;
        const int slot = (tid >> 4) << 1;            // 0 for lanes 0-15, 2 for 16-31
        const float* bp = &sm[ch * NTHR + half * 128 + slot];

        v2f a; a[0] = 1.f; a[1] = 1.f;               // A = all ones (16x4)
        v8f acc = {0.f, 0.f, 0.f, 0.f, 0.f, 0.f, 0.f, 0.f};
        #pragma unroll
        for (int s = 0; s < 32; ++s) {
            v2f bv; bv[0] = bp[s * 4]; bv[1] = bp[s * 4 + 1];
            acc = __builtin_amdgcn_wmma_f32_16x16x4_f32(
                false, a, false, bv, (short)0, acc, false, false);
        }
        float col = acc[0];                          // colsum[n] (rows all equal)
        float tot = col + __shfl_xor(col, 8, 32);    // fold the two 128-thread halves
        if (tid < 8)
            partials[(long long)blockIdx.x * 8 + tid] = tot;
    }
}

// Deterministic final reduction over per-block partials + scalar loss.
__global__ __launch_bounds__(NTHR)
void det_loss_final(const float* __restrict__ partials, float* __restrict__ out,
                    int nblk, float batch) {
    __shared__ float sm[NTHR];
    const int tid = threadIdx.x;
    const int ch = tid & 7, slot = tid >> 3;         // 8 channels x 32 slots
    float s = 0.f;
    for (int b = slot; b < nblk; b += 32)            // fixed order -> deterministic
        s += partials[(long long)b * 8 + ch];
    sm[tid] = s;
    __syncthreads();
    for (int off = 128; off >= 8; off >>= 1) {       // keeps channel alignment (off % 8 == 0)
        if (tid < off) sm[tid] += sm[tid + off];
        __syncthreads();
    }
    if (tid == 0) {
        float ceS = sm[0];
        float TN = sm[1], FP = sm[2], FN = sm[3], TP = sm[4];
        float ce     = ceS / batch;
        float meanCS = FN / batch;                   // M[0][1]=1 is the only nonzero cost
        bool  nz     = (TP > 0.f) && (TN > 0.f) && (FP > 0.f) && (FN > 0.f);
        float ratio  = (TP / fmaxf(TP + FN, 1.f)) * (FP / fmaxf(FP + TN, 1.f));
        float coeff  = nz ? (-LAMBD * logf(sqrtf(fmaxf(ratio, 1e-30f)))) : LAMBD;
        out[0] = ce + coeff * meanCS;
    }
}

extern "C" void kernel_launch(void* const* d_in, const int* in_sizes, int n_in,
                              void* d_out, int out_size, void* d_ws, size_t ws_size,
                              hipStream_t stream) {
    (void)n_in; (void)out_size;
    const float4* out4 = (const float4*)d_in[0];     // outputs [B,2] f32 -> pairs of items
    const int2*   lab2 = (const int2*)d_in[1];       // labels  [B]  i32 -> pairs
    float* ws  = (float*)d_ws;
    float* out = (float*)d_out;

    const long long batch = (long long)in_sizes[1];  // element count of labels
    const long long pairs = batch / 2;

    int nblk = 2048;                                 // 2048 blk x 256 thr = 0.5M threads
    size_t need = (size_t)nblk * 8 * sizeof(float);
    if (ws_size < need) {
        nblk = (int)(ws_size / (8 * sizeof(float)));
        if (nblk < 1) nblk = 1;
    }

    det_loss_main<<<nblk, NTHR, 0, stream>>>(out4, lab2, ws, pairs);
    det_loss_final<<<1, NTHR, 0, stream>>>(ws, out, nblk, (float)batch);
}